// DetectionPostprocess_6700148982162
// MI455X (gfx1250) — compile-verified
//
#include <hip/hip_runtime.h>
#include <stdint.h>

#define D3      48
#define NANCH   (D3*D3*D3)          // 110592
#define TOPK    60
#define CHUNK   4096
#define NCHUNK  (NANCH/CHUNK)       // 27
#define KTOT    (2*TOPK)            // 120
#define NMSK    20
#define THRESH      0.15f
#define NMS_THRESH  0.05f
#define STRIDEF     2.0f            // 96/48 in all three dims

#if defined(__has_builtin)
  #if __has_builtin(__builtin_amdgcn_global_load_async_to_lds_b128)
    #define HAVE_ASYNC_LDS 1
  #endif
  #if __has_builtin(__builtin_amdgcn_s_wait_asynccnt)
    #define HAVE_WAIT_ASYNC 1
  #endif
#endif

typedef unsigned long long u64;
typedef unsigned int u32;
typedef int v4i __attribute__((ext_vector_type(4)));
typedef __attribute__((address_space(1))) v4i g_v4i;   // global
typedef __attribute__((address_space(3))) v4i l_v4i;   // LDS

__device__ __forceinline__ float sigmoid_xla(float x) {
    // XLA lowers logistic as 0.5 + 0.5*tanh(0.5*x)
    return 0.5f + 0.5f * tanhf(0.5f * x);
}

__device__ __forceinline__ u64 pack_key(float score, u32 idx) {
    return ((u64)__float_as_uint(score) << 32) | (u64)(0xFFFFFFFFu - idx);
}

__device__ __forceinline__ void wait_async_lds() {
#ifdef HAVE_WAIT_ASYNC
    __builtin_amdgcn_s_wait_asynccnt(0);
#else
    asm volatile("s_wait_asynccnt 0" ::: "memory");
#endif
}

// ---------------------------------------------------------------------------
// Stage A: per-(batch,head,chunk) top-60 over 4096 sigmoid scores.
// grid = B*2*NCHUNK blocks, 256 threads.
// ---------------------------------------------------------------------------
__global__ __launch_bounds__(256) void topk_chunk_kernel(
    const float* __restrict__ cls0, const float* __restrict__ cls1,
    u64* __restrict__ partials)
{
    __shared__ __align__(16) float vals[CHUNK];   // 16 KB
    __shared__ u64 keys[CHUNK];                   // 32 KB
    __shared__ u64 red[256];

    const int blk   = blockIdx.x;
    const int chunk = blk % NCHUNK;
    const int bh    = blk / NCHUNK;
    const int head  = bh & 1;
    const int b     = bh >> 1;
    const int t     = threadIdx.x;

    const float* __restrict__ cls = head ? cls1 : cls0;
    const float* __restrict__ src = cls + (size_t)b * NANCH + (size_t)chunk * CHUNK;

    // ---- stage 4096 floats into LDS (CDNA5 async global->LDS if available)
#ifdef HAVE_ASYNC_LDS
    {
        #pragma unroll
        for (int j = 0; j < 4; ++j) {
            int f = t + 256 * j;
            g_v4i* gp = (g_v4i*)((const v4i*)src + f);
            l_v4i* lp = (l_v4i*)((v4i*)vals + f);
            __builtin_amdgcn_global_load_async_to_lds_b128(gp, lp, 0, 0);
        }
        wait_async_lds();
    }
#else
    {
        const float4* src4 = (const float4*)src;
        float4* dst4 = (float4*)vals;
        #pragma unroll
        for (int j = 0; j < 4; ++j) {
            int f = t + 256 * j;
            dst4[f] = src4[f];
        }
    }
#endif
    __syncthreads();

    // ---- build sortable keys: (sigmoid bits << 32) | ~global_index
    #pragma unroll
    for (int j = 0; j < 16; ++j) {
        int i = t + 256 * j;
        float s = sigmoid_xla(vals[i]);
        u32 gidx = (u32)(chunk * CHUNK + i);
        keys[i] = pack_key(s, gidx);
    }
    __syncthreads();

    // ---- 60 rounds of block-wide argmax with invalidation
    u64* __restrict__ outp = partials + (size_t)blk * TOPK;
    for (int r = 0; r < TOPK; ++r) {
        u64 m = 0;
        #pragma unroll
        for (int j = 0; j < 16; ++j) {
            u64 k = keys[t + 256 * j];
            m = (k > m) ? k : m;
        }
        red[t] = m;
        __syncthreads();
        for (int off = 128; off > 0; off >>= 1) {
            if (t < off) { u64 o = red[t + off]; if (o > red[t]) red[t] = o; }
            __syncthreads();
        }
        u64 win = red[0];
        if (t == 0) outp[r] = win;
        // invalidate winner (keys are unique & nonzero)
        #pragma unroll
        for (int j = 0; j < 16; ++j) {
            int i = t + 256 * j;
            if (keys[i] == win) keys[i] = 0ull;
        }
        __syncthreads();
    }
}

// ---------------------------------------------------------------------------
// Stage B: merge 27 partial top-60 lists -> top-60 per (batch,head); decode
// boxes by gathering offset/shape only at the 60 winning indices.
// grid = B*2 blocks, 256 threads.
// ---------------------------------------------------------------------------
__global__ __launch_bounds__(256) void topk_merge_kernel(
    const u64* __restrict__ partials,
    const float* __restrict__ shp0, const float* __restrict__ off0,
    const float* __restrict__ shp1, const float* __restrict__ off1,
    float* __restrict__ cand)
{
    __shared__ u64 keys[NCHUNK * TOPK];   // 1620 * 8 B
    __shared__ u64 red[256];
    __shared__ u64 winners[TOPK];

    const int bh   = blockIdx.x;
    const int head = bh & 1;
    const int b    = bh >> 1;
    const int t    = threadIdx.x;
    const int NCAND = NCHUNK * TOPK;

    const u64* __restrict__ src = partials + (size_t)bh * NCAND;
    for (int i = t; i < NCAND; i += 256) keys[i] = src[i];
    __syncthreads();

    for (int r = 0; r < TOPK; ++r) {
        u64 m = 0;
        for (int i = t; i < NCAND; i += 256) { u64 k = keys[i]; if (k > m) m = k; }
        red[t] = m;
        __syncthreads();
        for (int off = 128; off > 0; off >>= 1) {
            if (t < off) { u64 o = red[t + off]; if (o > red[t]) red[t] = o; }
            __syncthreads();
        }
        u64 win = red[0];
        if (t == 0) winners[r] = win;
        for (int i = t; i < NCAND; i += 256) if (keys[i] == win) keys[i] = 0ull;
        __syncthreads();
    }

    // ---- decode + sparse gather: 60 threads in parallel
    if (t < TOPK) {
        u64 k = winners[t];
        float s  = __uint_as_float((u32)(k >> 32));
        u32 gidx = 0xFFFFFFFFu - (u32)(k & 0xFFFFFFFFu);
        int z   = (int)(gidx / (D3 * D3));
        int rem = (int)(gidx % (D3 * D3));
        int y   = rem / D3;
        int x   = rem % D3;

        const float* __restrict__ shp = head ? shp1 : shp0;
        const float* __restrict__ off = head ? off1 : off0;
        size_t base = (size_t)b * 3 * NANCH + gidx;
        float oz = off[base];
        float oy = off[base + (size_t)NANCH];
        float ox = off[base + (size_t)(2 * NANCH)];
        float dz = 2.0f * shp[base];
        float dy = 2.0f * shp[base + (size_t)NANCH];
        float dx = 2.0f * shp[base + (size_t)(2 * NANCH)];

        float* __restrict__ c = cand + ((size_t)b * KTOT + head * TOPK + t) * 8;
        c[0] = s;
        c[1] = ((float)z + oz) * STRIDEF;
        c[2] = ((float)y + oy) * STRIDEF;
        c[3] = ((float)x + ox) * STRIDEF;
        c[4] = dz; c[5] = dy; c[6] = dx;
        c[7] = 0.0f;
    }
}

// ---------------------------------------------------------------------------
// Stage C: per-batch sort(desc, index tie-break) + greedy NMS + compaction.
// grid = B blocks, 128 threads.
// ---------------------------------------------------------------------------
__global__ __launch_bounds__(128) void nms_kernel(
    const float* __restrict__ cand, float* __restrict__ out)
{
    __shared__ float sraw[KTOT];
    __shared__ float ss[KTOT];
    __shared__ float bb[KTOT][6];
    __shared__ int   keepf[NMSK];
    __shared__ int   posf[NMSK];

    const int b = blockIdx.x;
    const int t = threadIdx.x;
    const float* __restrict__ cb = cand + (size_t)b * KTOT * 8;

    if (t < KTOT) sraw[t] = cb[t * 8];
    __syncthreads();

    // rank sort by (score desc, original position asc); invalid (low) scores
    // naturally land at the back, same effect as the reference's -inf trick
    if (t < KTOT) {
        u64 mykey = pack_key(sraw[t], (u32)t);
        int rank = 0;
        for (int j = 0; j < KTOT; ++j) {
            u64 kj = pack_key(sraw[j], (u32)j);
            if (kj > mykey) rank++;
        }
        ss[rank] = sraw[t];
        #pragma unroll
        for (int c = 0; c < 6; ++c) bb[rank][c] = cb[t * 8 + 1 + c];
    }
    __syncthreads();

    if (t == 0) {
        bool keep[NMSK];
        for (int i = 0; i < NMSK; ++i) {
            bool ci = ss[i] > THRESH;
            bool sup = false;
            if (ci) {
                float ci_lo[3], ci_hi[3], vi = 1.0f;
                #pragma unroll
                for (int c = 0; c < 3; ++c) {
                    float ctr = bb[i][c], sz = bb[i][3 + c];
                    ci_lo[c] = ctr - 0.5f * sz;
                    ci_hi[c] = ctr + 0.5f * sz;
                    vi *= sz;
                }
                for (int j = 0; j < i && !sup; ++j) {
                    if (!keep[j]) continue;
                    float inter = 1.0f, vj = 1.0f;
                    #pragma unroll
                    for (int c = 0; c < 3; ++c) {
                        float ctr = bb[j][c], sz = bb[j][3 + c];
                        float lo = ctr - 0.5f * sz;
                        float hi = ctr + 0.5f * sz;
                        float lo2 = ci_lo[c] > lo ? ci_lo[c] : lo;
                        float hi2 = ci_hi[c] < hi ? ci_hi[c] : hi;
                        float d = hi2 - lo2;
                        inter *= (d > 0.0f) ? d : 0.0f;
                        vj *= sz;
                    }
                    float iou = inter / (vi + vj - inter);
                    if (iou > NMS_THRESH) sup = true;
                }
            }
            keep[i] = ci && !sup;
            keepf[i] = keep[i] ? 1 : 0;
        }
        int p = 0;
        for (int i = 0; i < NMSK; ++i) { posf[i] = p; if (keepf[i]) p++; }
    }
    __syncthreads();

    float* __restrict__ ob = out + (size_t)b * KTOT * 8;
    for (int i = t; i < KTOT * 8; i += 128) ob[i] = -1.0f;
    __syncthreads();

    if (t < NMSK && keepf[t]) {
        float* r = ob + (size_t)posf[t] * 8;
        r[0] = 1.0f;
        r[1] = ss[t];
        #pragma unroll
        for (int c = 0; c < 6; ++c) r[2 + c] = bb[t][c];
    }
}

// ---------------------------------------------------------------------------
extern "C" void kernel_launch(void* const* d_in, const int* in_sizes, int n_in,
                              void* d_out, int out_size, void* d_ws, size_t ws_size,
                              hipStream_t stream)
{
    const float* cls1 = (const float*)d_in[0];
    const float* shp1 = (const float*)d_in[1];
    const float* off1 = (const float*)d_in[2];
    const float* cls2 = (const float*)d_in[3];
    const float* shp2 = (const float*)d_in[4];
    const float* off2 = (const float*)d_in[5];

    const int B = in_sizes[0] / NANCH;   // 32

    u64* partials = (u64*)d_ws;
    size_t partial_bytes = (size_t)B * 2 * NCHUNK * TOPK * sizeof(u64);
    float* cand = (float*)((char*)d_ws + ((partial_bytes + 255) & ~(size_t)255));

    topk_chunk_kernel<<<B * 2 * NCHUNK, 256, 0, stream>>>(cls1, cls2, partials);
    topk_merge_kernel<<<B * 2, 256, 0, stream>>>(partials, shp1, off1, shp2, off2, cand);
    nms_kernel<<<B, 128, 0, stream>>>(cand, (float*)d_out);
}